// NonMaximumSuppress_87634512707766
// MI455X (gfx1250) — compile-verified
//
#include <hip/hip_runtime.h>
#include <stdint.h>

// SSD box decode for MI455X (gfx1250).
// Memory-bound elementwise op: use the Tensor Data Mover to DMA the strided
// 4-of-85-channel coordinate tiles into LDS, then compute + B128 stores.

#define TILE   1024          // anchors per workgroup (all levels divide this)
#define BLOCK  256           // 8 wave32s
#define PER_THREAD (TILE / BLOCK)

#if defined(__has_builtin)
#  if __has_builtin(__builtin_amdgcn_tensor_load_to_lds) && \
      __has_builtin(__builtin_amdgcn_s_wait_tensorcnt)
#    define HAS_TDM 1
#  endif
#endif
#ifndef HAS_TDM
#  define HAS_TDM 0
#endif

typedef unsigned int v4u __attribute__((ext_vector_type(4)));
typedef int          v8i __attribute__((ext_vector_type(8)));
typedef int          v4i __attribute__((ext_vector_type(4)));

// sqrt(ar) and 1/sqrt(ar) for ASPECT_RATIOS = {1, 2, 0.5, 3, 0.3333333, 1.2}
__constant__ float c_sq[6] = {1.0f, 1.41421356f, 0.70710678f,
                              1.73205081f, 0.57735025f, 1.09544512f};
__constant__ float c_rq[6] = {1.0f, 0.70710678f, 1.41421356f,
                              0.57735027f, 1.73205090f, 0.91287093f};

__global__ __launch_bounds__(BLOCK)
void ssd_decode_kernel(const float* __restrict__ src,   // (B,h,w,6,85)
                       float* __restrict__ dst,         // (B,h,w,6,4)
                       int wShift,                      // log2(w) == log2(h)
                       float s512,                      // scale * 512
                       float cS,                        // 512 / w  (== 512 / h)
                       float sxy)                       // w / 512  (== h / 512)
{
  __shared__ __align__(16) float lds[TILE * 4];
  const int start = (int)blockIdx.x * TILE;

#if HAS_TDM
  if (threadIdx.x < 32) {   // wave 0 issues the tensor DMA (EXEC ignored by TDM)
    const uint64_t ga  = (uint64_t)(uintptr_t)src + (uint64_t)start * 340ull; // 85 floats/row
    const uint32_t lof = (uint32_t)(uintptr_t)(void*)lds; // low 32b of generic = LDS byte offset

    v4u g0;
    g0.x = 1u;                                            // count = 1 valid descriptor
    g0.y = lof;                                           // lds_addr (bytes)
    g0.z = (uint32_t)ga;                                  // global_addr[31:0]
    g0.w = ((uint32_t)(ga >> 32) & 0x01FFFFFFu)           // global_addr[56:32]
         | 0x80000000u;                                   // type = 2 ("image")

    v8i g1;
    g1[0] = 0x00020000;           // workgroup_mask=0 (not in cluster), data_size=4B
    g1[1] = (int)(85u  << 16);    // atomic_barrier_addr=0 | tensor_dim0 = 85
    g1[2] = (int)((unsigned)TILE << 16); // tensor_dim0 hi=0 | tensor_dim1 = TILE
    g1[3] = (int)(4u   << 16);    // tensor_dim1 hi=0 | tile_dim0 = 4 elements (16B)
    g1[4] = TILE;                 // tile_dim1 = TILE rows, tile_dim2 = 0 (2D tile)
    g1[5] = 85;                   // tensor_dim0_stride = 85 elements (lo32)
    g1[6] = 0;                    // stride hi16 = 0, tensor_dim1_stride = 0
    g1[7] = 0;

    v4i g2 = {0, 0, 0, 0};        // 2D tensor: groups 2/3 unused
    v4i g3 = {0, 0, 0, 0};
    v8i g4 = {0, 0, 0, 0, 0, 0, 0, 0}; // trailing group (clang-23 6-arg form), unused for 2D

    __builtin_amdgcn_tensor_load_to_lds(g0, g1, g2, g3, g4, 0);
    __builtin_amdgcn_s_wait_tensorcnt(0);                 // TENSORcnt == 0
  }
  __syncthreads();
#endif

#pragma unroll
  for (int k = 0; k < PER_THREAD; ++k) {
    const int local = (int)threadIdx.x + k * BLOCK;
    const int n     = start + local;                      // flat anchor index (< 2^20)

    float t0, t1, t2, t3;
#if HAS_TDM
    const float4 t = ((const float4*)lds)[local];         // ds_load_b128
    t0 = t.x; t1 = t.y; t2 = t.z; t3 = t.w;
#else
    const float* p = src + (long long)n * 85;
    t0 = p[0]; t1 = p[1]; t2 = p[2]; t3 = p[3];
#endif

    const int a = n % 6;                                  // aspect-ratio index
    const int r = n / 6;
    const int m = (1 << wShift) - 1;
    const int x = r & m;
    const int y = (r >> wShift) & m;

    const float bw = s512 * c_sq[a];                      // s*512*sqrt(ar)
    const float bh = s512 * c_rq[a];                      // s*512/sqrt(ar)

    const float px = fmaf(bh, t0, ((float)x + 0.5f) * cS);   // cx + bh*t0
    const float py = fmaf(bw, t1, ((float)y + 0.5f) * cS);   // cy + bw*t1
    const float pw = bw * __expf(t2);
    const float ph = bh * __expf(t3);

    float4 o;
    o.x = px * sxy;
    o.y = py * sxy;
    o.z = pw * sxy;
    o.w = ph * sxy;
    ((float4*)dst)[n] = o;                                // global_store_b128
  }
}

extern "C" void kernel_launch(void* const* d_in, const int* in_sizes, int n_in,
                              void* d_out, int out_size, void* d_ws, size_t ws_size,
                              hipStream_t stream) {
  (void)in_sizes; (void)n_in; (void)out_size; (void)d_ws; (void)ws_size;

  static const int   FS[5] = {64, 32, 16, 8, 4};
  static const int   SH[5] = {6, 5, 4, 3, 2};
  static const float SC[5] = {0.1f, 0.2f, 0.375f, 0.55f, 0.725f};

  long long off = 0;
  for (int i = 0; i < 5; ++i) {
    const int h       = FS[i];
    const int anchors = 32 * h * h * 6;        // all are exact multiples of TILE
    const int grid    = anchors / TILE;
    ssd_decode_kernel<<<grid, BLOCK, 0, stream>>>(
        (const float*)d_in[i],
        (float*)d_out + off,
        SH[i],
        SC[i] * 512.0f,
        512.0f / (float)h,
        (float)h / 512.0f);
    off += (long long)anchors * 4;
  }
}